// InvariantEncoding_18820546691340
// MI455X (gfx1250) — compile-verified
//
#include <hip/hip_runtime.h>

typedef __attribute__((ext_vector_type(2))) float v2f;
typedef __attribute__((ext_vector_type(8))) float v8f;

#define N_ELEC  1024
#define N_ATOMS 64
#define N_UP    512
#define D_ONE   256
#define D_TWO   32
#define OUT_FEAT (D_ONE + 2 * D_TWO) /* 320 */

__device__ __forceinline__ v8f wmma16x16x4(v2f a, v2f b, v8f c) {
    // D = A(16x4) * B(4x16) + C ;  (neg_a, A, neg_b, B, c_mod, C, reuse_a, reuse_b)
    return __builtin_amdgcn_wmma_f32_16x16x4_f32(false, a, false, b, (short)0, c, false, false);
}

// gfx1250 has V_TANH_F32 (TRANS32); fall back to branch-free exp2+rcp.
__device__ __forceinline__ float fast_tanh(float x) {
#if __has_builtin(__builtin_amdgcn_tanhf)
    return __builtin_amdgcn_tanhf(x);
#else
    const float e = __builtin_amdgcn_exp2f(x * 2.8853900817779268f);
    return 1.0f - 2.0f * __builtin_amdgcn_rcpf(e + 1.0f);
#endif
}

__device__ __forceinline__ float fast_sqrt(float x) {
#if __has_builtin(__builtin_amdgcn_sqrtf)
    return __builtin_amdgcn_sqrtf(x);
#else
    return sqrtf(x);
#endif
}

// One spin segment of the e-e sweep: j in [jbeg, jend), 16-row WMMA tiles.
// Uniform control flow throughout (EXEC all-1s at the WMMAs).
__device__ __forceinline__ void ee_segment(
    int jbeg, int jend, const float* __restrict__ se,
    float eix, float eiy, float eiz, bool hi, int half,
    v2f bA, v2f bB, v8f cA, v8f cB, float& accA, float& accB)
{
    for (int j0 = jbeg; j0 < jend; j0 += 16) {
        const int j = j0 + half;               // row M = half (both lane halves)
        const float dx = eix - se[3 * j + 0];
        const float dy = eiy - se[3 * j + 1];
        const float dz = eiz - se[3 * j + 2];
        const float nr = fast_sqrt(dx * dx + dy * dy + dz * dz); // j==i -> all-zero feature
        v2f a;
        a.x = hi ? dz : dx;                    // lanes 0-15: (K0,K1)=(dx,dy); 16-31: (K2,K3)=(dz,|r|)
        a.y = hi ? nr : dy;

        const v8f dA = wmma16x16x4(a, bA, cA); // channels 0..15
        const v8f dB = wmma16x16x4(a, bB, cB); // channels 16..31

        float sA = 0.f, sB = 0.f;
        for (int v = 0; v < 8; ++v) { sA += fast_tanh(dA[v]); sB += fast_tanh(dB[v]); }
        accA += sA;
        accB += sB;
    }
}

// ---------------------------------------------------------------------------
// Kernel A: e-e pairwise features -> Dense(4->32) -> tanh -> spin-segmented
// mean -> write h_two columns [256,320) tiled over the 64 atom slots.
// One wave per electron i; electrons staged in LDS; up/dn segments split so
// the inner loop has no per-iteration selection.
// ---------------------------------------------------------------------------
__global__ __launch_bounds__(256) void ie_ktwo_kernel(
    const float* __restrict__ electrons,   // (1024,3)
    const float* __restrict__ W_two,       // (4,32) row-major
    const float* __restrict__ b_two,       // (32,)
    float* __restrict__ out)               // (1024,64,320) region
{
    __shared__ float se[3 * N_ELEC];       // 12 KB electron table
    for (int t = threadIdx.x; t < 3 * N_ELEC; t += 256) se[t] = electrons[t];
    __syncthreads();

    const int lane = threadIdx.x & 31;
    const int wave = threadIdx.x >> 5;
    const int i    = blockIdx.x * 8 + wave;      // 128 blocks * 8 waves = 1024
    const int half = lane & 15;
    const bool hi  = lane >= 16;

    const float eix = se[3 * i + 0];
    const float eiy = se[3 * i + 1];
    const float eiz = se[3 * i + 2];

    // B operand: VGPR0 = {K=0 | K=2}, VGPR1 = {K=1 | K=3}, N = lane%16 (+16 for 2nd tile)
    const int k0 = hi ? 2 : 0;
    v2f bA, bB;
    bA.x = W_two[k0 * D_TWO + half];
    bA.y = W_two[(k0 + 1) * D_TWO + half];
    bB.x = W_two[k0 * D_TWO + 16 + half];
    bB.y = W_two[(k0 + 1) * D_TWO + 16 + half];

    // Bias folded into C: lane L holds column N = L%16 in all 8 C VGPRs
    v8f cA, cB;
    {
        const float bvA = b_two[half];
        const float bvB = b_two[16 + half];
        for (int v = 0; v < 8; ++v) { cA[v] = bvA; cB[v] = bvB; }
    }

    float accA_up = 0.f, accA_dn = 0.f, accB_up = 0.f, accB_dn = 0.f;
    ee_segment(0,    N_UP,   se, eix, eiy, eiz, hi, half, bA, bB, cA, cB, accA_up, accB_up);
    ee_segment(N_UP, N_ELEC, se, eix, eiy, eiz, hi, half, bA, bB, cA, cB, accA_dn, accB_dn);

    // rows 0-7 live in lanes 0-15, rows 8-15 in lanes 16-31 -> fold halves
    accA_up += __shfl_xor(accA_up, 16, 32);
    accA_dn += __shfl_xor(accA_dn, 16, 32);
    accB_up += __shfl_xor(accB_up, 16, 32);
    accB_dn += __shfl_xor(accB_dn, 16, 32);

    const float mu = (hi ? accB_up : accA_up) * (1.0f / N_UP);
    const float md = (hi ? accB_dn : accA_dn) * (1.0f / (N_ELEC - N_UP));
    const int  ch  = half + (hi ? 16 : 0);     // each lane owns one of 32 channels

    float* base = out + (size_t)i * N_ATOMS * OUT_FEAT;
    for (int m = 0; m < N_ATOMS; ++m) {
        base[m * OUT_FEAT + D_ONE + ch]         = mu;   // cols 256..287 (up mean)
        base[m * OUT_FEAT + D_ONE + D_TWO + ch] = md;   // cols 288..319 (dn mean)
    }
}

// ---------------------------------------------------------------------------
// Kernel B: e-n features -> (+nuc emb) -> Dense(4->256) -> tanh -> h_one
// columns [0,256), plus raw r_im4 into the second output region.
// One wave per (electron i, 16-atom tile); N=256 swept in 16 WMMA tiles.
// ---------------------------------------------------------------------------
__global__ __launch_bounds__(256) void ie_kone_kernel(
    const float* __restrict__ electrons,   // (1024,3)
    const float* __restrict__ atoms,       // (64,3)
    const float* __restrict__ W_one,       // (4,256) row-major
    const float* __restrict__ b_one,       // (256,)
    const float* __restrict__ nuc,         // (1,64,4)
    float* __restrict__ out)
{
    const int lane = threadIdx.x & 31;
    const int wave = threadIdx.x >> 5;
    const int gw   = blockIdx.x * 8 + wave;    // 512 blocks * 8 waves = 4096
    const int i    = gw >> 2;
    const int m0   = (gw & 3) * 16;
    const int half = lane & 15;
    const bool hi  = lane >= 16;
    const int m    = m0 + half;

    const float dx = electrons[3 * i + 0] - atoms[3 * m + 0];
    const float dy = electrons[3 * i + 1] - atoms[3 * m + 1];
    const float dz = electrons[3 * i + 2] - atoms[3 * m + 2];
    const float nr = fast_sqrt(dx * dx + dy * dy + dz * dz);

    const float fx = dx + nuc[4 * m + 0];
    const float fy = dy + nuc[4 * m + 1];
    const float fz = dz + nuc[4 * m + 2];
    const float fn = nr + nuc[4 * m + 3];

    v2f a;
    a.x = hi ? fz : fx;
    a.y = hi ? fn : fy;

    const int k0    = hi ? 2 : 0;
    const int rbase = hi ? 8 : 0;              // D rows owned by this lane half
    const size_t rowStart = (size_t)i * N_ATOMS + m0;

    for (int n0 = 0; n0 < D_ONE; n0 += 16) {
        v2f b;
        b.x = W_one[k0 * D_ONE + n0 + half];
        b.y = W_one[(k0 + 1) * D_ONE + n0 + half];
        v8f c;
        const float bv = b_one[n0 + half];
        for (int v = 0; v < 8; ++v) c[v] = bv;

        const v8f d = wmma16x16x4(a, b, c);
        for (int v = 0; v < 8; ++v) {
            out[(rowStart + rbase + v) * OUT_FEAT + n0 + half] = fast_tanh(d[v]);
        }
    }

    // r_im4 region (after all WMMAs; divergence here is safe)
    if (!hi) {
        float* r2 = out + (size_t)N_ELEC * N_ATOMS * OUT_FEAT;
        const size_t off = ((size_t)i * N_ATOMS + m) * 4;
        r2[off + 0] = dx; r2[off + 1] = dy; r2[off + 2] = dz; r2[off + 3] = nr;
    }
}

extern "C" void kernel_launch(void* const* d_in, const int* in_sizes, int n_in,
                              void* d_out, int out_size, void* d_ws, size_t ws_size,
                              hipStream_t stream) {
    const float* electrons = (const float*)d_in[0];  // (1024,3)
    const float* atoms     = (const float*)d_in[1];  // (64,3)
    const float* W_two     = (const float*)d_in[2];  // (4,32)
    const float* b_two     = (const float*)d_in[3];  // (32,)
    const float* W_one     = (const float*)d_in[4];  // (4,256)
    const float* b_one     = (const float*)d_in[5];  // (256,)
    const float* nuc       = (const float*)d_in[6];  // (1,64,4)
    float* out = (float*)d_out;

    (void)in_sizes; (void)n_in; (void)out_size; (void)d_ws; (void)ws_size;

    // e-e path: 1024 waves (one per electron), 8 waves/block
    ie_ktwo_kernel<<<128, 256, 0, stream>>>(electrons, W_two, b_two, out);
    // e-n path: 4096 waves (one per electron x 16-atom tile), 8 waves/block
    ie_kone_kernel<<<512, 256, 0, stream>>>(electrons, atoms, W_one, b_one, nuc, out);
}